// ContrastiveLoss_51273319580309
// MI455X (gfx1250) — compile-verified
//
#include <hip/hip_runtime.h>

#define MARGIN 200.0f
#define EPS 1e-6f

typedef __attribute__((ext_vector_type(2))) float v2f;
typedef __attribute__((ext_vector_type(8))) float v8f;

// ---------------------------------------------------------------------------
// K0: sq[r] = sum_d feats[r,d]^2   (rows >= 64*200 use (x - EPS)^2, i.e. ap)
// One block per (row,t) pair; 2048 floats per row; coalesced float4 loads.
// ---------------------------------------------------------------------------
__global__ void k0_rownorm(const float* __restrict__ feats, float* __restrict__ sq) {
    const int bx  = blockIdx.x;                 // 0..25599  (= row*200 + t flat)
    const int tid = threadIdx.x;                // 256 threads
    const float* row = feats + (size_t)bx * 2048;
    const bool isA = bx >= 64 * 200;

    float s = 0.0f;
    #pragma unroll
    for (int i = 0; i < 2; ++i) {
        float4 v = *(const float4*)(row + (size_t)(tid + 256 * i) * 4);
        if (isA) { v.x -= EPS; v.y -= EPS; v.z -= EPS; v.w -= EPS; }
        s += v.x * v.x + v.y * v.y + v.z * v.z + v.w * v.w;
    }
    __shared__ float red[256];
    red[tid] = s;
    __syncthreads();
    for (int off = 128; off; off >>= 1) {
        if (tid < off) red[tid] += red[tid + off];
        __syncthreads();
    }
    if (tid == 0) sq[bx] = red[0];
}

// ---------------------------------------------------------------------------
// K1: per-t 64x64 GEMM over d=2048 via V_WMMA_F32_16X16X4_F32, then hinge
// term, reduced over the 8 waves (8 t's) of the block into partial[tc][i][j].
// grid = (16 tiles, 25 t-chunks), block = 256 threads (8 waves, 1 t each).
// ---------------------------------------------------------------------------
__global__ void __launch_bounds__(256)
k1_cross(const float* __restrict__ feats, const float* __restrict__ sq,
         float* __restrict__ partial) {
    __shared__ float smem[16384];               // 64 KB: 8 waves * (1K A + 1K B)

    const int tid  = threadIdx.x;
    const int w    = tid >> 5;                  // wave 0..7
    const int lane = tid & 31;
    const int t    = blockIdx.y * 8 + w;        // 0..199 (25*8 == 200 exactly)
    const int tile = blockIdx.x;                // 0..15
    const int i0   = (tile >> 2) * 16;          // n-row tile base
    const int j0   = (tile & 3) * 16;           // a-row tile base

    float* As = smem + w * 2048;                // 16 x 64 f32
    float* Bs = As + 1024;                      // 16 x 64 f32

    const int m  = lane & 15;                   // fragment row / B column
    const int kh = lane >> 4;                   // K-half select (K0,1 vs K2,3)

    v8f c = {};                                 // 16x16 f32 accumulator

    for (int kc = 0; kc < 2048; kc += 64) {
        // ---- stage A (n rows) and B (a rows, EPS folded) tiles into LDS ----
        #pragma unroll
        for (int s = 0; s < 8; ++s) {
            int idx = s * 32 + lane;            // 0..255 float4 slots
            int r   = idx >> 4;                 // 0..15 row
            int c4  = (idx & 15) * 4;           // column (floats)
            float4 an = *(const float4*)(feats +
                ((size_t)(i0 + r) * 200 + t) * 2048 + kc + c4);
            float4 aa = *(const float4*)(feats +
                ((size_t)(64 + j0 + r) * 200 + t) * 2048 + kc + c4);
            aa.x -= EPS; aa.y -= EPS; aa.z -= EPS; aa.w -= EPS;
            *(float4*)(As + r * 64 + c4) = an;
            *(float4*)(Bs + r * 64 + c4) = aa;
        }
        __syncthreads();

        // ---- 16 WMMA steps: K=64 chunk in K=4 slices ----
        #pragma unroll
        for (int k2 = 0; k2 < 64; k2 += 4) {
            v2f af = *(const v2f*)(As + m * 64 + k2 + 2 * kh);
            v2f bf = *(const v2f*)(Bs + m * 64 + k2 + 2 * kh);
            c = __builtin_amdgcn_wmma_f32_16x16x4_f32(
                    false, af, false, bf, (short)0, c, false, false);
        }
        __syncthreads();
    }

    // ---- hinge: contrib = max(MARGIN - sqrt(max(n2+a2-2c,0)),0)^2 ----
    const float a2v = sq[(size_t)(64 + j0 + m) * 200 + t];   // column norm
    float contrib[8];
    #pragma unroll
    for (int r = 0; r < 8; ++r) {
        int mr = r + 8 * kh;                                  // C-layout row
        float n2v   = sq[(size_t)(i0 + mr) * 200 + t];
        float dist2 = fmaxf(n2v + a2v - 2.0f * c[r], 0.0f);
        float u     = fmaxf(MARGIN - sqrtf(dist2), 0.0f);
        contrib[r]  = u * u;
    }

    // ---- reduce the 8 waves' 16x16 tiles, write one partial per thread ----
    #pragma unroll
    for (int r = 0; r < 8; ++r)
        smem[w * 256 + (r + 8 * kh) * 16 + m] = contrib[r];
    __syncthreads();

    float ssum = 0.0f;
    #pragma unroll
    for (int w2 = 0; w2 < 8; ++w2) ssum += smem[w2 * 256 + tid];
    partial[(size_t)blockIdx.y * 4096 + (i0 + (tid >> 4)) * 64 + j0 + (tid & 15)] = ssum;
}

// ---------------------------------------------------------------------------
// K2: d[i,j] = (1/200) * sum_tc partial; argmin (first-index ties), store
// loss_con = 0.001*d_min and (max_n, max_a). Single block, fixed-order sums.
// ---------------------------------------------------------------------------
__global__ void k2_argmin(const float* __restrict__ partial,
                          float* __restrict__ scal, int* __restrict__ idxs) {
    const int tid = threadIdx.x;                // 256
    float bm = __builtin_inff();
    int   bi = 0;
    for (int e = tid; e < 4096; e += 256) {
        float s = 0.0f;
        for (int tc = 0; tc < 25; ++tc) s += partial[(size_t)tc * 4096 + e];
        s *= (1.0f / 200.0f);
        if (s < bm) { bm = s; bi = e; }         // strict '<' => first min
    }
    __shared__ float rv[256];
    __shared__ int   ri[256];
    rv[tid] = bm; ri[tid] = bi;
    __syncthreads();
    for (int off = 128; off; off >>= 1) {
        if (tid < off) {
            float ov = rv[tid + off]; int oi = ri[tid + off];
            if (ov < rv[tid] || (ov == rv[tid] && oi < ri[tid])) {
                rv[tid] = ov; ri[tid] = oi;
            }
        }
        __syncthreads();
    }
    if (tid == 0) {
        scal[0] = 0.001f * rv[0];
        idxs[0] = ri[0] >> 6;                   // max_n
        idxs[1] = ri[0] & 63;                   // max_a
    }
}

// ---------------------------------------------------------------------------
// K3: dn[i] / da[j] vs. selected rows (indices read device-side from ws).
// 128 blocks (64 n-side + 64 a-side), streaming float4 over 200*2048 elems.
// ---------------------------------------------------------------------------
__global__ void k3_dev(const float* __restrict__ feats,
                       const int* __restrict__ idxs, float* __restrict__ dvec) {
    const int b    = blockIdx.x;                // 0..127
    const bool isA = b >= 64;
    const int  i   = isA ? b - 64 : b;
    const int  ref = isA ? idxs[1] : idxs[0];
    const int  base = isA ? 64 : 0;
    const float* src = feats + (size_t)(base + i)   * 409600;
    const float* rf  = feats + (size_t)(base + ref) * 409600;

    float s = 0.0f;
    if (i != ref) {
        for (int p = threadIdx.x; p < 102400; p += 256) {
            float4 x = ((const float4*)src)[p];
            float4 mref = ((const float4*)rf)[p];
            float dx = x.x - mref.x + EPS;
            float dy = x.y - mref.y + EPS;
            float dz = x.z - mref.z + EPS;
            float dw = x.w - mref.w + EPS;
            s += dx * dx + dy * dy + dz * dz + dw * dw;
        }
    }
    __shared__ float red[256];
    red[threadIdx.x] = s;
    __syncthreads();
    for (int off = 128; off; off >>= 1) {
        if (threadIdx.x < off) red[threadIdx.x] += red[threadIdx.x + off];
        __syncthreads();
    }
    if (threadIdx.x == 0) dvec[b] = red[0] * (1.0f / 200.0f);
}

// ---------------------------------------------------------------------------
// K4: out = loss_con + sum(dn)/64 + sum(da)/64   (rows matching ref are 0)
// ---------------------------------------------------------------------------
__global__ void k4_final(const float* __restrict__ dvec,
                         const float* __restrict__ scal, float* __restrict__ out) {
    const int tid = threadIdx.x;                // 128
    __shared__ float red[128];
    red[tid] = dvec[tid] * (1.0f / 64.0f);
    __syncthreads();
    for (int off = 64; off; off >>= 1) {
        if (tid < off) red[tid] += red[tid + off];
        __syncthreads();
    }
    if (tid == 0) out[0] = scal[0] + red[0];
}

// ---------------------------------------------------------------------------
extern "C" void kernel_launch(void* const* d_in, const int* in_sizes, int n_in,
                              void* d_out, int out_size, void* d_ws, size_t ws_size,
                              hipStream_t stream) {
    const float* feats = (const float*)d_in[0];
    float* out = (float*)d_out;

    float* ws      = (float*)d_ws;
    float* sq      = ws;                        // 25600 floats
    float* partial = ws + 25600;                // 25*4096 = 102400 floats
    float* dvec    = ws + 128000;               // 128 floats
    float* scal    = ws + 128128;               // 1 float
    int*   idxs    = (int*)(ws + 128129);       // 2 ints

    k0_rownorm<<<25600, 256, 0, stream>>>(feats, sq);
    k1_cross<<<dim3(16, 25), 256, 0, stream>>>(feats, sq, partial);
    k2_argmin<<<1, 256, 0, stream>>>(partial, scal, idxs);
    k3_dev<<<128, 256, 0, stream>>>(feats, idxs, dvec);
    k4_final<<<1, 128, 0, stream>>>(dvec, scal, out);
}